// LLSA_37701222924673
// MI455X (gfx1250) — compile-verified
//
#include <hip/hip_runtime.h>
#include <hip/hip_bf16.h>
#include <math.h>

typedef __attribute__((ext_vector_type(2))) float v2f;
typedef __attribute__((ext_vector_type(4))) float v4f;
typedef __attribute__((ext_vector_type(8))) float v8f;

#define SEG_S 8192
#define SEG_B 16
#define SEG_F 256
#define SEG_EPS 1e-5f
#define ZSLOTS 16

// ---------------------------------------------------------------------------
// Kernel 1: boundary scan. One block, 1024 threads, 8 positions each.
//   ind[s] = (s>0) && any_b(cp[b,s]!=0);  seg_id = inclusive cumsum(ind)
// Emits seg_start[g] for g in [0, nseg], with seg_start[nseg] = S sentinel,
// and nseg = seg_id[S-1] + 1.
// ---------------------------------------------------------------------------
__global__ __launch_bounds__(1024) void LLSA_seg_boundaries(
    const int* __restrict__ cp, int* __restrict__ nseg_out,
    int* __restrict__ seg_start) {
  __shared__ int lds[1024];
  constexpr int PER_T = SEG_S / 1024;  // 8
  const int t = threadIdx.x;
  const int base = t * PER_T;

  int ind[PER_T];
  int localsum = 0;
#pragma unroll
  for (int i = 0; i < PER_T; ++i) {
    const int s = base + i;
    int v = 0;
    if (s > 0) {
#pragma unroll
      for (int b = 0; b < SEG_B; ++b) v |= (cp[b * SEG_S + s] != 0) ? 1 : 0;
    }
    ind[i] = v;
    localsum += v;
  }

  lds[t] = localsum;
  __syncthreads();
  // Hillis-Steele inclusive scan across the 1024 per-thread sums.
  for (int off = 1; off < 1024; off <<= 1) {
    int v = (t >= off) ? lds[t - off] : 0;
    __syncthreads();
    lds[t] += v;
    __syncthreads();
  }
  int run = lds[t] - localsum;  // exclusive prefix for this thread's range

#pragma unroll
  for (int i = 0; i < PER_T; ++i) {
    const int s = base + i;
    run += ind[i];                      // inclusive seg_id at position s
    if (s == 0) seg_start[0] = 0;       // segment 0 always starts at 0
    if (ind[i]) seg_start[run] = s;     // unique boundary per segment id >= 1
    if (s == SEG_S - 1) {
      nseg_out[0] = run + 1;
      seg_start[run + 1] = SEG_S;       // sentinel end
    }
  }
}

// ---------------------------------------------------------------------------
// Kernel 2: one wave32 per (f-tile of 16, batch, segment-slot).
// Pass 1: WMMA-accumulated Sum(x) / Sum(x^2) with A = ones(16x4):
//   D[m,n] = sum_k B[k,n] + C[m,n]  -> every C row holds per-feature sums.
//   x loads are regular-temporal so the lines stay in L2 for pass 2.
// Pass 2: stats redistributed via wave32 shuffles so each lane owns 4
// consecutive features -> b128 load/store, 8 rows per wave-iteration.
//   x loads and out stores are non-temporal (last-use / never re-read) so
//   pass-2 traffic doesn't evict the x working set from L2.
// ---------------------------------------------------------------------------
__global__ __launch_bounds__(32) void LLSA_seg_norm(
    const float* __restrict__ x, const float* __restrict__ w,
    const float* __restrict__ bi, float* __restrict__ out,
    const int* __restrict__ nseg_in, const int* __restrict__ seg_start) {
  const int lane = threadIdx.x;
  const int half = lane >> 4;    // 0: K rows 0..1, 1: K rows 2..3
  const int fcol = lane & 15;    // column N of the WMMA tile
  const int fbase = blockIdx.x * 16;
  const int b = blockIdx.y;
  const int z = blockIdx.z;

  const float wv = w[fbase + fcol];
  const float bv = bi[fbase + fcol];
  const float* xb = x + (size_t)b * SEG_S * SEG_F;
  float* ob = out + (size_t)b * SEG_S * SEG_F;

  // Pass-2 layout: lane handles features [qf, qf+4) of this tile, row = lane>>2.
  const int qf = (lane & 3) * 4;
  const int rowoff = lane >> 2;  // 0..7
  const float w0 = __shfl(wv, qf), w1 = __shfl(wv, qf + 1),
              w2 = __shfl(wv, qf + 2), w3 = __shfl(wv, qf + 3);
  const float b0 = __shfl(bv, qf), b1 = __shfl(bv, qf + 1),
              b2 = __shfl(bv, qf + 2), b3 = __shfl(bv, qf + 3);

  const int nseg = __builtin_amdgcn_readfirstlane(nseg_in[0]);
  const v2f aones = {1.0f, 1.0f};  // A = ones(16x4): D accumulates col-sums of B

  for (int g = z; g < nseg; g += ZSLOTS) {
    const int s0 = __builtin_amdgcn_readfirstlane(seg_start[g]);
    const int s1 = __builtin_amdgcn_readfirstlane(seg_start[g + 1]);

    v8f acc = {};   // Sum(x)
    v8f acc2 = {};  // Sum(x*x)

    int s = s0;
    // Main loop: 32 s-rows per iteration = 8 WMMA chunk pairs,
    // 16 outstanding b32 loads per lane for latency hiding.
    for (; s + 32 <= s1; s += 32) {
#pragma unroll
      for (int c = 0; c < 8; ++c) {
        const int r = s + c * 4 + half * 2;
        const float* p = xb + (size_t)r * SEG_F + fbase + fcol;
        v2f bm;
        bm.x = p[0];      // K row (half*2)
        bm.y = p[SEG_F];  // K row (half*2 + 1)
        v2f bq = bm * bm;
        acc = __builtin_amdgcn_wmma_f32_16x16x4_f32(false, aones, false, bm,
                                                    (short)0, acc, false, false);
        acc2 = __builtin_amdgcn_wmma_f32_16x16x4_f32(false, aones, false, bq,
                                                     (short)0, acc2, false, false);
      }
    }
    // Masked tail: zero-padding is exact for Sum / Sum-of-squares.
    // Also keeps rows >= s1 exec-masked (last segment ends at S).
    for (; s < s1; s += 4) {
      const int r0 = s + half * 2;
      const int r1 = r0 + 1;
      v2f bm;
      bm.x = (r0 < s1) ? xb[(size_t)r0 * SEG_F + fbase + fcol] : 0.0f;
      bm.y = (r1 < s1) ? xb[(size_t)r1 * SEG_F + fbase + fcol] : 0.0f;
      v2f bq = bm * bm;
      acc = __builtin_amdgcn_wmma_f32_16x16x4_f32(false, aones, false, bm,
                                                  (short)0, acc, false, false);
      acc2 = __builtin_amdgcn_wmma_f32_16x16x4_f32(false, aones, false, bq,
                                                   (short)0, acc2, false, false);
    }

    // Every D row is identical; C VGPR0 at this lane = sums for feature fcol.
    const float ssum = acc[0];
    const float ssq = acc2[0];
    const float rc = 1.0f / (float)(s1 - s0);
    const float mean = ssum * rc;
    float var = ssq * rc - mean * mean;
    var = var > 0.0f ? var : 0.0f;
    const float inv = 1.0f / sqrtf(var + SEG_EPS);

    // Redistribute stats: lane needs mean/inv for features qf..qf+3
    // (stats live replicated in lanes 0..15, index = feature column).
    const float m0 = __shfl(mean, qf), m1 = __shfl(mean, qf + 1),
                m2 = __shfl(mean, qf + 2), m3 = __shfl(mean, qf + 3);
    const float i0 = __shfl(inv, qf), i1 = __shfl(inv, qf + 1),
                i2 = __shfl(inv, qf + 2), i3 = __shfl(inv, qf + 3);

    // Pass 2: b128 per lane, 4 lanes = 64B contiguous row slice, 8 rows/iter.
    // Non-temporal on both sides: x line is dead after this read; out is
    // write-once. Keeps the pass-1 x working set resident in L2.
    for (int ss = s0 + rowoff; ss < s1; ss += 8) {
      const size_t idx = (size_t)ss * SEG_F + fbase + qf;
      v4f v = __builtin_nontemporal_load((const v4f*)(xb + idx));
      v4f r;
      r.x = (v.x - m0) * i0 * w0 + b0;
      r.y = (v.y - m1) * i1 * w1 + b1;
      r.z = (v.z - m2) * i2 * w2 + b2;
      r.w = (v.w - m3) * i3 * w3 + b3;
      __builtin_nontemporal_store(r, (v4f*)(ob + idx));
    }
  }
}

extern "C" void kernel_launch(void* const* d_in, const int* in_sizes, int n_in,
                              void* d_out, int out_size, void* d_ws,
                              size_t ws_size, hipStream_t stream) {
  const float* x = (const float*)d_in[0];          // [B,S,F]
  const float* w = (const float*)d_in[1];          // [F]
  const float* bi = (const float*)d_in[2];         // [F]
  const int* cp = (const int*)d_in[3];             // [B,S]
  float* out = (float*)d_out;                      // [B,S,F]

  int* nseg = (int*)d_ws;                          // 1 int (16B slot)
  int* seg_start = (int*)((char*)d_ws + 16);       // S+1 ints

  LLSA_seg_boundaries<<<1, 1024, 0, stream>>>(cp, nseg, seg_start);

  dim3 grid(SEG_F / 16, SEG_B, ZSLOTS);
  LLSA_seg_norm<<<grid, 32, 0, stream>>>(x, w, bi, out, nseg, seg_start);
}